// GaussianModel_31671088841483
// MI455X (gfx1250) — compile-verified
//
#include <hip/hip_runtime.h>
#include <math.h>

typedef float v2f __attribute__((ext_vector_type(2)));
typedef float v8f __attribute__((ext_vector_type(8)));

// Projection pipeline for 3D gaussians.
// Per wave (32 lanes): each lane owns one gaussian; view transform R*(p-t)
// is done on the matrix pipe with V_WMMA_F32_16X16X4_F32 (two issues per
// wave, one per half-wave), everything else is streaming elementwise work
// with non-temporal load/store hints (working set >> 192MB L2).
__global__ __launch_bounds__(256) void gs_project_wmma(
    const float* __restrict__ pos,
    const float* __restrict__ scl,
    const float* __restrict__ col,
    const float* __restrict__ opa,
    const float* __restrict__ vm,
    const int*  __restrict__ dw,
    const int*  __restrict__ dh,
    float* __restrict__ out,
    int N)
{
    const int tid  = blockIdx.x * blockDim.x + threadIdx.x;
    const int lane = threadIdx.x & 31;
    const int row  = lane & 15;
    const int g    = (tid < N) ? tid : (N - 1);   // clamp => EXEC all-ones at WMMA

    // ---- A matrix (16x4, broadcast): rows 0..2 = [ R | -R*t ], rest zero ----
    // lane L<16 supplies A[L][0..1]; lane L+16 supplies A[L][2..3].
    v2f a; a[0] = 0.0f; a[1] = 0.0f;
    if (row < 3) {
        const float r0 = vm[4*row + 0], r1 = vm[4*row + 1], r2 = vm[4*row + 2];
        const float t0 = vm[3], t1 = vm[7], t2 = vm[11];
        if (lane < 16) { a[0] = r0; a[1] = r1; }
        else           { a[0] = r2; a[1] = -(r0*t0 + r1*t1 + r2*t2); }
    }

    // ---- per-lane gaussian position (streaming NT loads) ----
    const size_t g3 = 3 * (size_t)g;
    const float px = __builtin_nontemporal_load(pos + g3 + 0);
    const float py = __builtin_nontemporal_load(pos + g3 + 1);
    const float pz = __builtin_nontemporal_load(pos + g3 + 2);

    // cross-half copies (partner lane = lane ^ 16) to marshal B matrices
    const float pxo = __shfl_xor(px, 16, 32);
    const float pyo = __shfl_xor(py, 16, 32);
    const float pzo = __shfl_xor(pz, 16, 32);

    // B1 (4x16): columns = gaussians of lanes 0..15
    // B2 (4x16): columns = gaussians of lanes 16..31
    // lane L<16 holds B[0][L],B[1][L]; lane L+16 holds B[2][L],B[3][L].
    v2f b1, b2;
    if (lane < 16) { b1[0] = px;  b1[1] = py;   b2[0] = pxo; b2[1] = pyo; }
    else           { b1[0] = pzo; b1[1] = 1.0f; b2[0] = pz;  b2[1] = 1.0f; }

    v8f c; for (int i = 0; i < 8; ++i) c[i] = 0.0f;
    v8f d1 = __builtin_amdgcn_wmma_f32_16x16x4_f32(false, a, false, b1,
                                                   (short)0, c, false, false);
    v8f d2 = __builtin_amdgcn_wmma_f32_16x16x4_f32(false, a, false, b2,
                                                   (short)0, c, false, false);

    // D[m][n] lives in VGPR m, lane n. Lane L<16: own coords in d1[0..2].
    // Lane L+16: its coords sit at lane L of d2[0..2] -> shuffle back.
    const float x2 = __shfl_xor(d2[0], 16, 32);
    const float y2 = __shfl_xor(d2[1], 16, 32);
    const float z2 = __shfl_xor(d2[2], 16, 32);

    float xv = (lane < 16) ? d1[0] : x2;
    float yv = (lane < 16) ? d1[1] : y2;
    float zv = (lane < 16) ? d1[2] : z2;

    // pos_view = clip(..., -100, 100)
    xv = fminf(fmaxf(xv, -100.0f), 100.0f);
    yv = fminf(fmaxf(yv, -100.0f), 100.0f);
    zv = fminf(fmaxf(zv, -100.0f), 100.0f);

    const float W = (float)(*dw);
    const float H = (float)(*dh);

    const float valid = (zv > 0.1f && zv < 10.0f) ? 1.0f : 0.0f;
    const float z = fminf(fmaxf(zv, 0.1f), 10.0f);

    const float xp = fminf(fmaxf(xv * 500.0f / z + 0.5f * W, -1000.0f), W + 1000.0f);
    const float yp = fminf(fmaxf(-(yv * 500.0f) / z + 0.5f * H, -1000.0f), H + 1000.0f);

    const float s0 = __builtin_nontemporal_load(scl + g3 + 0);
    const float s1 = __builtin_nontemporal_load(scl + g3 + 1);
    const float sx = fmaxf(s0, 0.001f) * 500.0f / z;
    const float sy = fmaxf(s1, 0.001f) * 500.0f / z;
    const float c00 = fminf(fmaxf(sx * sx + 1e-4f, 1e-6f), 1e6f);
    const float c11 = fminf(fmaxf(sy * sy + 1e-4f, 1e-6f), 1e6f);

    const float cr = fminf(fmaxf(__builtin_nontemporal_load(col + g3 + 0), 0.0f), 1.0f);
    const float cg = fminf(fmaxf(__builtin_nontemporal_load(col + g3 + 1), 0.0f), 1.0f);
    const float cb = fminf(fmaxf(__builtin_nontemporal_load(col + g3 + 2), 0.0f), 1.0f);

    const float o   = __builtin_nontemporal_load(opa + (size_t)g);
    const float sig = 1.0f / (1.0f + expf(-o));

    if (tid < N) {
        const size_t n = (size_t)N;
        const size_t t = (size_t)tid;
        float* proj = out;            // (N,3)
        float* cov  = out + 3 * n;    // (N,2,2) -> 4 per gaussian
        float* co   = out + 7 * n;    // (N,3)
        float* op   = out + 10 * n;   // (N,)
        float* mk   = out + 11 * n;   // (N,) mask as 0/1 float

        __builtin_nontemporal_store(xp, proj + 3 * t + 0);
        __builtin_nontemporal_store(yp, proj + 3 * t + 1);
        __builtin_nontemporal_store(zv, proj + 3 * t + 2);

        __builtin_nontemporal_store(c00,   cov + 4 * t + 0);
        __builtin_nontemporal_store(1e-6f, cov + 4 * t + 1);
        __builtin_nontemporal_store(1e-6f, cov + 4 * t + 2);
        __builtin_nontemporal_store(c11,   cov + 4 * t + 3);

        __builtin_nontemporal_store(cr, co + 3 * t + 0);
        __builtin_nontemporal_store(cg, co + 3 * t + 1);
        __builtin_nontemporal_store(cb, co + 3 * t + 2);

        __builtin_nontemporal_store(sig,   op + t);
        __builtin_nontemporal_store(valid, mk + t);
    }
}

extern "C" void kernel_launch(void* const* d_in, const int* in_sizes, int n_in,
                              void* d_out, int out_size, void* d_ws, size_t ws_size,
                              hipStream_t stream) {
    // setup_inputs() order:
    // 0: positions (N,3) f32   1: scales (N,3) f32   2: rotations (N,4) f32 (unused)
    // 3: colors (N,3) f32      4: opacities (N,) f32 5: viewmat (4,4) f32
    // 6: projmat (4,4) f32 (unused)                  7: width (int)  8: height (int)
    const float* pos = (const float*)d_in[0];
    const float* scl = (const float*)d_in[1];
    const float* col = (const float*)d_in[3];
    const float* opa = (const float*)d_in[4];
    const float* vm  = (const float*)d_in[5];
    const int*   dw  = (const int*)d_in[7];
    const int*   dh  = (const int*)d_in[8];

    const int N = in_sizes[4];                 // opacities count == N_GAUSS
    const int threads = 256;
    const int blocks  = (N + threads - 1) / threads;

    gs_project_wmma<<<blocks, threads, 0, stream>>>(
        pos, scl, col, opa, vm, dw, dh, (float*)d_out, N);
}